// ISDLoss_only_type1_17489106829328
// MI455X (gfx1250) — compile-verified
//
#include <hip/hip_runtime.h>
#include <hip/hip_bf16.h>
#include <stdint.h>

// ISD loss (masked symmetric KL) for [B=128, N=8732, C=21] f32.
// Memory-bound: 282 MB streamed once -> ~12 us floor at 23.3 TB/s.
// CDNA5 path: async global->LDS staging (ASYNCcnt, b128 wide), s_wait_asynccnt,
// wave32 shuffle reduction, deterministic 2-stage final reduce.

#define BDIM 128
#define NPRI 8732
#define NCLS 21
#define HALF 64
#define EPS_F 1e-7f

#define ROWS (BDIM * NPRI)            // 1,117,696
#define TPB 256                       // threads per block (8 wave32)
#define ROWS_PER_BLK TPB              // one row per thread
#define NBLK (ROWS / ROWS_PER_BLK)    // 4366 (exact)
#define TILE_ELEMS (ROWS_PER_BLK * NCLS)  // 5376 floats per tensor tile
#define TILE_F4 (TILE_ELEMS / 4)          // 1344 float4
#define F4_ROUNDS (TILE_F4 / TPB)         // 5 full b128 rounds (1280 float4)
#define TAIL_BASE (F4_ROUNDS * TPB * 4)   // 5120; tail = 256 floats (1 b32/thread)

static_assert(ROWS % ROWS_PER_BLK == 0, "rows must tile exactly");
static_assert(TILE_ELEMS - TAIL_BASE == TPB, "tail must be exactly one b32 per thread");

// ---- CDNA5 async global->LDS copies (ASYNCcnt-tracked, no VGPR bounce) ----
__device__ __forceinline__ void async_g2l_b32(uint32_t lds_off, const float* gaddr) {
    asm volatile("global_load_async_to_lds_b32 %0, %1, off"
                 :: "v"(lds_off), "v"((uint64_t)(uintptr_t)gaddr)
                 : "memory");
}

__device__ __forceinline__ void async_g2l_b128(uint32_t lds_off, const float* gaddr) {
    // 16B-aligned on both sides; 32 lanes -> 512 contiguous bytes per issue.
    asm volatile("global_load_async_to_lds_b128 %0, %1, off"
                 :: "v"(lds_off), "v"((uint64_t)(uintptr_t)gaddr)
                 : "memory");
}

__device__ __forceinline__ void wait_async0() {
#if __has_builtin(__builtin_amdgcn_s_wait_asynccnt)
    __builtin_amdgcn_s_wait_asynccnt(0);
#else
    asm volatile("s_wait_asynccnt 0x0" ::: "memory");
#endif
}

__device__ __forceinline__ uint32_t lds_addr_of(const void* p) {
    // Shared-memory generic pointers carry the wave-relative LDS offset in
    // their low 32 bits (LDS aperture: addr[31:0] = LDS offset).
    return (uint32_t)(uintptr_t)p;
}

__global__ __launch_bounds__(TPB) void isd_partial_kernel(
    const float* __restrict__ lam_p,
    const float* __restrict__ conf,
    const float* __restrict__ shuf,
    const float* __restrict__ interp,
    float* __restrict__ partial)      // [2 * NBLK] (sum, cnt) pairs
{
    __shared__ alignas(16) float sA[TILE_ELEMS];  // conf rows
    __shared__ alignas(16) float sB[TILE_ELEMS];  // half-swapped shuffle rows
    __shared__ alignas(16) float sC[TILE_ELEMS];  // interpolation rows
    __shared__ float red[2 * (TPB / 32)];

    const uint32_t t   = threadIdx.x;
    const uint32_t blk = blockIdx.x;
    const uint32_t r0  = blk * ROWS_PER_BLK;          // first row of this tile
    const uint32_t cb  = r0 * NCLS;                   // flat element base

    // ---- conf / interp: wide b128 streaming copies (16B aligned tile base) ----
    #pragma unroll
    for (uint32_t k = 0; k < F4_ROUNDS; ++k) {
        const uint32_t e = (t + k * TPB) * 4;         // float index, 16B aligned
        async_g2l_b128(lds_addr_of(&sA[e]), conf   + cb + e);
        async_g2l_b128(lds_addr_of(&sC[e]), interp + cb + e);
    }
    {   // 256-float tail, one dword per thread
        const uint32_t e = TAIL_BASE + t;
        async_g2l_b32(lds_addr_of(&sA[e]), conf   + cb + e);
        async_g2l_b32(lds_addr_of(&sC[e]), interp + cb + e);
    }

    // ---- shuffle: half-swapped row gather (piecewise contiguous, 4B aligned) ----
    #pragma unroll
    for (uint32_t k = 0; k < TILE_ELEMS / TPB; ++k) {
        const uint32_t e   = t + k * TPB;             // 0..5375
        const uint32_t q   = e / NCLS;                // row within tile
        const uint32_t c   = e - q * NCLS;            // class
        const uint32_t row = r0 + q;
        const uint32_t b   = row / NPRI;
        const uint32_t n   = row - b * NPRI;
        const uint32_t tb  = (b + HALF) & (BDIM - 1);
        const uint32_t src = (tb * NPRI + n) * NCLS + c;
        async_g2l_b32(lds_addr_of(&sB[e]), shuf + src);
    }
    wait_async0();
    __syncthreads();

    // ---- Per-row fused KL + foreground mask (one row per lane) ----
    const float lam  = lam_p[0];
    const float lam1 = 1.0f - lam;
    const uint32_t base = t * NCLS;   // 21 coprime with 64 banks -> conflict-free

    const float x0 = sA[base];
    const float y0 = sB[base];
    float mx = -3.4e38f, my = -3.4e38f;
    float kl;
    {
        const float mixed = fmaf(lam, x0, fmaf(lam1, y0, EPS_F));
        const float inter = sC[base] + EPS_F;
        kl = (inter - mixed) * (__logf(inter) - __logf(mixed));
    }
    #pragma unroll
    for (uint32_t c = 1; c < NCLS; ++c) {
        const float x = sA[base + c];
        const float y = sB[base + c];
        const float z = sC[base + c];
        mx = fmaxf(mx, x);
        my = fmaxf(my, y);
        const float m2 = fmaf(lam, x, fmaf(lam1, y, EPS_F));
        const float i2 = z + EPS_F;
        kl += (i2 - m2) * (__logf(i2) - __logf(m2));
    }
    const bool sel = (mx > x0) && (my > y0);
    float v  = sel ? kl : 0.0f;
    float cn = sel ? 1.0f : 0.0f;

    // ---- wave32 reduction ----
    #pragma unroll
    for (int off = 16; off > 0; off >>= 1) {
        v  += __shfl_down(v,  off, 32);
        cn += __shfl_down(cn, off, 32);
    }
    if ((t & 31u) == 0u) {
        red[2 * (t >> 5) + 0] = v;
        red[2 * (t >> 5) + 1] = cn;
    }
    __syncthreads();

    if (t == 0) {
        float vv = 0.0f, cc = 0.0f;
        #pragma unroll
        for (int w = 0; w < TPB / 32; ++w) { vv += red[2 * w]; cc += red[2 * w + 1]; }
        partial[2 * blk + 0] = vv;
        partial[2 * blk + 1] = cc;
    }
}

__global__ __launch_bounds__(TPB) void isd_final_kernel(
    const float* __restrict__ partial, float* __restrict__ out)
{
    __shared__ float ssum[TPB];
    __shared__ float scnt[TPB];
    const uint32_t t = threadIdx.x;
    float s = 0.0f, c = 0.0f;
    for (uint32_t i = t; i < NBLK; i += TPB) {   // fixed order -> deterministic
        s += partial[2 * i + 0];
        c += partial[2 * i + 1];
    }
    ssum[t] = s; scnt[t] = c;
    __syncthreads();
    #pragma unroll
    for (int off = TPB / 2; off > 0; off >>= 1) {
        if (t < (uint32_t)off) { ssum[t] += ssum[t + off]; scnt[t] += scnt[t + off]; }
        __syncthreads();
    }
    if (t == 0) {
        const float cnt  = scnt[0];
        const float loss = ssum[0] / fmaxf(cnt, 1.0f);
        out[0] = (cnt > 0.0f) ? (loss * 0.5f) : 0.0f;
    }
}

extern "C" void kernel_launch(void* const* d_in, const int* in_sizes, int n_in,
                              void* d_out, int out_size, void* d_ws, size_t ws_size,
                              hipStream_t stream) {
    const float* lam    = (const float*)d_in[0];  // scalar
    const float* conf   = (const float*)d_in[1];  // [128, 8732, 21]
    const float* shuf   = (const float*)d_in[2];  // [128, 8732, 21]
    const float* interp = (const float*)d_in[3];  // [128, 8732, 21]
    float* out     = (float*)d_out;
    float* partial = (float*)d_ws;                // needs 2*NBLK*4 = 34,928 B

    isd_partial_kernel<<<NBLK, TPB, 0, stream>>>(lam, conf, shuf, interp, partial);
    isd_final_kernel<<<1, TPB, 0, stream>>>(partial, out);
}